// MS_39496519254112
// MI455X (gfx1250) — compile-verified
//
#include <hip/hip_runtime.h>

typedef __attribute__((ext_vector_type(2))) float v2f;
typedef __attribute__((ext_vector_type(8))) float v8f;

#define NPIX   9216          // 96*96
#define DEMB   32
#define NTILES (NPIX / 16)   // 576
#define NWAVES 4
#define XSTRIDE 20           // padded LDS row stride for x tile (16B aligned rows)
#define ESTRIDE 17           // padded LDS row stride for E tile
#define BW     3.0f
#define STEP   0.5f
#define LOG2E  1.44269504088896340736f

// One block = 4 waves. Each block owns a 16-column j-tile; the 4 waves split the
// i-stream (interleaved) and reduce partial Y / d through LDS at the end.
__global__ __launch_bounds__(32 * NWAVES)
void ms_iter_kernel(const float* __restrict__ xin,
                    float* __restrict__ xout,
                    float* __restrict__ xout2)
{
    __shared__ __align__(16) float xtile[NWAVES][DEMB * XSTRIDE]; // 32x16 x-slab per wave
    __shared__ float etile[NWAVES][16 * ESTRIDE];                 // 16x16 E tile per wave
    __shared__ float red[NWAVES][32][17];                         // y0[8], y1[8], dacc

    const int lane  = threadIdx.x & 31;
    const int wave  = threadIdx.x >> 5;
    const int half  = lane >> 4;    // 0: lanes 0-15, 1: lanes 16-31
    const int l15   = lane & 15;
    const int jbase = blockIdx.x * 16;

    float* xt = xtile[wave];
    float* et = etile[wave];

    // ---- Preload GEMM1 B operands: the fixed j-tile of x -------------------
    // B layout (4x16, K=d rows): reg0 -> rows k0+{0,2}, reg1 -> rows k0+{1,3}
    v2f bj[8];
#pragma unroll
    for (int k0 = 0; k0 < 8; ++k0) {
        const int d0 = 4 * k0 + 2 * half;
        bj[k0].x = xin[(size_t)d0 * NPIX + jbase + l15];
        bj[k0].y = xin[(size_t)(d0 + 1) * NPIX + jbase + l15];
    }

    v8f y0 = {};   // Y rows d = 0..15
    v8f y1 = {};   // Y rows d = 16..31
    float dacc = 0.0f;

    for (int it = wave; it < NTILES; it += NWAVES) {
        const int ibase = it * 16;

        // ---- Stage x[:, ibase:ibase+16] into LDS (coalesced float4) --------
        {
            const float4* src = (const float4*)(xin + (size_t)lane * NPIX + ibase);
            float4* dst = (float4*)(xt + lane * XSTRIDE);
            dst[0] = src[0]; dst[1] = src[1]; dst[2] = src[2]; dst[3] = src[3];
        }
        if (ibase + 16 * NWAVES < NPIX)
            __builtin_prefetch(xin + (size_t)lane * NPIX + ibase + 16 * NWAVES, 0, 0);

        // ---- GEMM1: S[16i x 16j] = x_i^T x_j, K = 32 via 8 chained WMMAs ---
        v8f s = {};
#pragma unroll
        for (int k0 = 0; k0 < 8; ++k0) {
            const int d0 = 4 * k0 + 2 * half;
            v2f a;                                  // A[m=i, k=d] = xtile[d][i]
            a.x = xt[d0 * XSTRIDE + l15];
            a.y = xt[(d0 + 1) * XSTRIDE + l15];
            s = __builtin_amdgcn_wmma_f32_16x16x4_f32(false, a, false, bj[k0],
                                                      (short)0, s, false, false);
        }

        // ---- E = exp(BW*S) on VALU; column partial sums; stash E to LDS ----
        float esum = 0.0f;
#pragma unroll
        for (int v = 0; v < 8; ++v) {
            const float e = __builtin_amdgcn_exp2f(s[v] * (BW * LOG2E));
            esum += e;
            et[(v + 8 * half) * ESTRIDE + l15] = e;   // C-layout -> (i, j) tile
        }
        dacc += esum;

        // ---- GEMM2: Y[32d x 16j] += x_tile[32d x 16i] * E[16i x 16j] -------
#pragma unroll
        for (int c = 0; c < 4; ++c) {
            const int k0 = 4 * c + 2 * half;
            v2f b2;                                 // B[k=i, n=j] from E tile
            b2.x = et[k0 * ESTRIDE + l15];
            b2.y = et[(k0 + 1) * ESTRIDE + l15];

            v2f a2;                                 // A[m=d(0..15), k=i]
            a2.x = xt[l15 * XSTRIDE + k0];
            a2.y = xt[l15 * XSTRIDE + k0 + 1];
            y0 = __builtin_amdgcn_wmma_f32_16x16x4_f32(false, a2, false, b2,
                                                       (short)0, y0, false, false);

            v2f a3;                                 // A[m=d(16..31), k=i]
            a3.x = xt[(16 + l15) * XSTRIDE + k0];
            a3.y = xt[(16 + l15) * XSTRIDE + k0 + 1];
            y1 = __builtin_amdgcn_wmma_f32_16x16x4_f32(false, a3, false, b2,
                                                       (short)0, y1, false, false);
        }
    }

    // ---- Cross-wave reduction of partial Y and d through LDS ---------------
#pragma unroll
    for (int v = 0; v < 8; ++v) {
        red[wave][lane][v]     = y0[v];
        red[wave][lane][8 + v] = y1[v];
    }
    red[wave][lane][16] = dacc;
    __syncthreads();

    if (wave == 0) {
        v8f Y0 = {}, Y1 = {};
        float dtot = 0.0f;
        for (int w = 0; w < NWAVES; ++w) {
#pragma unroll
            for (int v = 0; v < 8; ++v) {
                Y0[v] += red[w][lane][v];
                Y1[v] += red[w][lane][8 + v];
            }
            dtot += red[w][lane][16];
        }
        // halves hold i-rows 0-7 / 8-15 of the same column j = l15: combine.
        const float dfull = dtot + __shfl_xor(dtot, 16, 32);
        const float sc = STEP / dfull;

#pragma unroll
        for (int v = 0; v < 8; ++v) {
            const int m = v + 8 * half;                // C-layout row within half
            const size_t j = (size_t)jbase + l15;
            const size_t o0 = (size_t)m * NPIX + j;
            const size_t o1 = (size_t)(16 + m) * NPIX + j;
            const float r0 = Y0[v] * sc + (1.0f - STEP) * xin[o0];
            const float r1 = Y1[v] * sc + (1.0f - STEP) * xin[o1];
            xout[o0] = r0;
            xout[o1] = r1;
            if (xout2) { xout2[o0] = r0; xout2[o1] = r1; }
        }
    }
}

extern "C" void kernel_launch(void* const* d_in, const int* in_sizes, int n_in,
                              void* d_out, int out_size, void* d_ws, size_t ws_size,
                              hipStream_t stream) {
    (void)in_sizes; (void)n_in; (void)d_ws; (void)ws_size; (void)out_size;
    const float* x0 = (const float*)d_in[0];
    float* out = (float*)d_out;
    const size_t F = (size_t)DEMB * NPIX;   // 294912 elements per frame

    // d_out layout: [final (F)] [outs[0] (F)] [outs[1] (F)] [outs[2] (F)]
    dim3 grid(NTILES), block(32 * NWAVES);
    // iter 0: x0 -> outs[0]
    ms_iter_kernel<<<grid, block, 0, stream>>>(x0, out + F, nullptr);
    // iter 1: outs[0] -> outs[1]
    ms_iter_kernel<<<grid, block, 0, stream>>>(out + F, out + 2 * F, nullptr);
    // iter 2: outs[1] -> outs[2] and final
    ms_iter_kernel<<<grid, block, 0, stream>>>(out + 2 * F, out + 3 * F, out);
}